// SSIM_86191403696744
// MI455X (gfx1250) — compile-verified
//
#include <hip/hip_runtime.h>

// ---------------------------------------------------------------------------
// SSIM over (32,3,512,512) fp32, 7x7 box window, interior crop, grand mean.
// Fused single-pass tile kernel:
//   vertical 7-tap  -> per-column prefix sums in LDS (1 global read / element)
//   horizontal 7-tap-> banded matmul on the matrix core:
//                      v_wmma_f32_16x16x4_f32 chained over K (22 -> pad 24)
//   SSIM formula + interior mask + deterministic block reduction.
// ---------------------------------------------------------------------------

typedef float v2f __attribute__((ext_vector_type(2)));
typedef float v8f __attribute__((ext_vector_type(8)));

#define IMG_H      512
#define IMG_W      512
#define N_PLANES   96          // 32 batch * 3 channels
#define TILE       16
#define HALO       22          // TILE + 6
#define PS_ROWS    23          // HALO rows of prefix + leading zero row
#define PS_COLS    24          // HALO cols padded to 24 (K pad, b64 alignment)
#define PLANE_F    (PS_ROWS * PS_COLS)      // 552 floats
#define WAVE_F     (5 * PLANE_F)            // 2760 floats per wave
#define WAVES_PB   4
#define THREADS_PB (WAVES_PB * 32)
#define TILES_X    (IMG_W / TILE)           // 32
#define TILES_Y    (IMG_H / TILE)           // 32
#define TILES_PER_PLANE (TILES_X * TILES_Y) // 1024
#define TOTAL_TILES (N_PLANES * TILES_PER_PLANE)      // 98304
#define NBLOCKS    (TOTAL_TILES / WAVES_PB)           // 24576
#define VALID_LO   3
#define VALID_HI   508

__device__ __forceinline__ v8f wmma_f32_16x16x4(v2f a, v2f b, v8f c) {
    // D = A(16x4,f32) * B(4x16,f32) + C(16x16,f32); 8-arg VOP3P form:
    // (neg_a, A, neg_b, B, c_mod, C, reuse_a, reuse_b)
    return __builtin_amdgcn_wmma_f32_16x16x4_f32(
        false, a, false, b, (short)0, c, false, false);
}

__global__ __launch_bounds__(THREADS_PB)
void ssim_stage1(const float* __restrict__ img1,
                 const float* __restrict__ img2,
                 float* __restrict__ part) {
    __shared__ float lds_ps[WAVES_PB * WAVE_F];   // prefix-sum planes
    __shared__ float lds_red[THREADS_PB];         // block reduction

    const int tid  = threadIdx.x;
    const int lane = tid & 31;
    const int wave = tid >> 5;

    const int tile  = blockIdx.x * WAVES_PB + wave;
    const int plane = tile >> 10;                 // / TILES_PER_PLANE
    const int rem   = tile & 1023;
    const int ty    = (rem >> 5) << 4;            // tile origin row
    const int tx    = (rem & 31) << 4;            // tile origin col

    const float* __restrict__ p1 = img1 + (size_t)plane * (IMG_H * IMG_W);
    const float* __restrict__ p2 = img2 + (size_t)plane * (IMG_H * IMG_W);
    float* ps = lds_ps + wave * WAVE_F;

    // ---- Phase 1: vertical prefix sums of {x, y, xx, yy, xy} ----
    // Lane c owns halo column c. PS[p][j][c] = sum of product rows 0..j-1.
    if (lane < 24) {
        if (lane < 22) {
            int col = tx - 3 + lane;
            col = col < 0 ? 0 : (col > IMG_W - 1 ? IMG_W - 1 : col);
            float a0 = 0.f, a1 = 0.f, a2 = 0.f, a3 = 0.f, a4 = 0.f;
            ps[0 * PLANE_F + lane] = 0.f;
            ps[1 * PLANE_F + lane] = 0.f;
            ps[2 * PLANE_F + lane] = 0.f;
            ps[3 * PLANE_F + lane] = 0.f;
            ps[4 * PLANE_F + lane] = 0.f;
            #pragma unroll 2
            for (int rr = 0; rr < HALO; ++rr) {
                int row = ty - 3 + rr;
                row = row < 0 ? 0 : (row > IMG_H - 1 ? IMG_H - 1 : row);
                const float x = p1[row * IMG_W + col];
                const float y = p2[row * IMG_W + col];
                a0 += x; a1 += y; a2 += x * x; a3 += y * y; a4 += x * y;
                const int o = (rr + 1) * PS_COLS + lane;
                ps[0 * PLANE_F + o] = a0;
                ps[1 * PLANE_F + o] = a1;
                ps[2 * PLANE_F + o] = a2;
                ps[3 * PLANE_F + o] = a3;
                ps[4 * PLANE_F + o] = a4;
            }
        } else {
            // zero the K-pad columns 22, 23 so padded WMMA chunks contribute 0
            for (int j = 0; j < PS_ROWS; ++j) {
                const int o = j * PS_COLS + lane;
                ps[0 * PLANE_F + o] = 0.f;
                ps[1 * PLANE_F + o] = 0.f;
                ps[2 * PLANE_F + o] = 0.f;
                ps[3 * PLANE_F + o] = 0.f;
                ps[4 * PLANE_F + o] = 0.f;
            }
        }
    }
    __syncthreads();

    // ---- Phase 2: horizontal 7-tap as banded matmul on the matrix core ----
    // A layout (32b, 16x4): lanes 0-15 = rows, VGPR{0,1} = K{0,1};
    //                       lanes 16-31 = rows, VGPR{0,1} = K{2,3}.
    // B layout (32b, 4x16): VGPR0 = K row {0|2} across lanes, VGPR1 = {1|3}.
    const int m    = lane & 15;            // A row == B column for this lane
    const int koff = (lane >> 4) << 1;     // 0 or 2

    v2f bmat[6];
    #pragma unroll
    for (int k = 0; k < 6; ++k) {
        const int kk = 4 * k + koff;       // global K of VGPR0 slot
        bmat[k].x = (kk     >= m && kk     <= m + 6) ? 1.0f : 0.0f;
        bmat[k].y = (kk + 1 >= m && kk + 1 <= m + 6) ? 1.0f : 0.0f;
    }

    v8f acc[5];
    #pragma unroll
    for (int p = 0; p < 5; ++p) {
        const float* pp = ps + p * PLANE_F;
        v8f c = {0.f, 0.f, 0.f, 0.f, 0.f, 0.f, 0.f, 0.f};
        #pragma unroll
        for (int k = 0; k < 6; ++k) {
            const int ci = 4 * k + koff;   // even -> 8B aligned b64 LDS loads
            const float2 hi = *(const float2*)(pp + (m + 7) * PS_COLS + ci);
            const float2 lo = *(const float2*)(pp +  m      * PS_COLS + ci);
            v2f a;                          // CS[m][k] = PS[m+7][k] - PS[m][k]
            a.x = hi.x - lo.x;
            a.y = hi.y - lo.y;
            c = wmma_f32_16x16x4(a, bmat[k], c);
        }
        acc[p] = c;
    }

    // ---- Phase 3: SSIM formula on accumulators + interior mask ----
    // D layout: VGPR g -> row (g | g+8), column = lane&15.
    const int rowoff = (lane >> 4) << 3;   // 0 or 8
    const float invNP = 1.0f / 49.0f;
    const float CN    = 49.0f / 48.0f;
    const float C1    = 1e-4f;             // (0.01)^2
    const float C2    = 9e-4f;             // (0.03)^2

    float lsum = 0.f;
    #pragma unroll
    for (int g = 0; g < 8; ++g) {
        const float ux  = acc[0][g] * invNP;
        const float uy  = acc[1][g] * invNP;
        const float uxx = acc[2][g] * invNP;
        const float uyy = acc[3][g] * invNP;
        const float uxy = acc[4][g] * invNP;
        const float vx  = CN * (uxx - ux * ux);
        const float vy  = CN * (uyy - uy * uy);
        const float vxy = CN * (uxy - ux * uy);
        const float A1  = 2.0f * ux * uy + C1;
        const float A2  = 2.0f * vxy + C2;
        const float B1  = ux * ux + uy * uy + C1;
        const float B2  = vx + vy + C2;
        const float S   = (A1 * A2) / (B1 * B2);
        const int gy = ty + g + rowoff;
        const int gx = tx + m;
        if (gy >= VALID_LO && gy <= VALID_HI && gx >= VALID_LO && gx <= VALID_HI)
            lsum += S;
    }

    // ---- Phase 4: deterministic block reduction -> one partial per block ----
    lds_red[tid] = lsum;
    __syncthreads();
    #pragma unroll
    for (int s = THREADS_PB / 2; s > 0; s >>= 1) {
        if (tid < s) lds_red[tid] += lds_red[tid + s];
        __syncthreads();
    }
    if (tid == 0) part[blockIdx.x] = lds_red[0];
}

__global__ __launch_bounds__(256)
void ssim_stage2(const float* __restrict__ part, float* __restrict__ out) {
    __shared__ double red[256];
    const int tid = threadIdx.x;
    double s = 0.0;
    for (int i = tid; i < NBLOCKS; i += 256) s += (double)part[i];
    red[tid] = s;
    __syncthreads();
    #pragma unroll
    for (int k = 128; k > 0; k >>= 1) {
        if (tid < k) red[tid] += red[tid + k];
        __syncthreads();
    }
    if (tid == 0)
        out[0] = (float)(red[0] / ((double)N_PLANES * 506.0 * 506.0));
}

extern "C" void kernel_launch(void* const* d_in, const int* in_sizes, int n_in,
                              void* d_out, int out_size, void* d_ws, size_t ws_size,
                              hipStream_t stream) {
    const float* img1 = (const float*)d_in[0];
    const float* img2 = (const float*)d_in[1];
    float* part = (float*)d_ws;            // NBLOCKS floats = 96 KB scratch
    float* out  = (float*)d_out;

    ssim_stage1<<<NBLOCKS, THREADS_PB, 0, stream>>>(img1, img2, part);
    ssim_stage2<<<1, 256, 0, stream>>>(part, out);
}